// GAT_48180943126853
// MI455X (gfx1250) — compile-verified
//
#include <hip/hip_runtime.h>
#include <math.h>

// ---------------- problem constants (from reference) ----------------
#define NN_   50000
#define EE_   800000
#define GG_   64
#define IN_   128
#define HID_  64
#define HEADS_ 4
#define OUT_  128
#define F1_   (HEADS_ * HID_)   // 256
#define EN_   (EE_ + NN_)       // 850000 edges incl. self loops
#define NEG_  0.2f

typedef __attribute__((ext_vector_type(2))) float v2f;
typedef __attribute__((ext_vector_type(8))) float v8f;

// ---------------- monotone float<->uint map for atomic max ----------------
__device__ __forceinline__ unsigned fmap(float f) {
    unsigned u = __float_as_uint(f);
    return (u & 0x80000000u) ? ~u : (u | 0x80000000u);
}
__device__ __forceinline__ float funmap(unsigned u) {
    unsigned v = (u & 0x80000000u) ? (u & 0x7FFFFFFFu) : ~u;
    return __uint_as_float(v);
}

// ---------------- fp32 WMMA GEMM: C[M,Nn] = A[M,K] @ B[K,Nn] ----------------
// One wave (32-thread block) computes a 16x64 strip: 4 accumulators share one
// A fragment per K-step -> 4 WMMAs per A load (4x A reuse vs 1 tile/wave).
// V_WMMA_F32_16X16X4_F32 fragment layout (wave32):
//   A (16x4): lane l holds row (l&15), K = k + 2*(l>>4) + {0,1}
//   B (4x16): lane l holds col (l&15), K = k + 2*(l>>4) + {0,1}
//   C/D     : VGPR v, lane l -> row = v + 8*(l>>4), col = (l&15)
__global__ void wmma_gemm_f32(const float* __restrict__ A,
                              const float* __restrict__ B,
                              float* __restrict__ C,
                              int M, int K, int Nn) {
    const int lane  = threadIdx.x & 31;
    const int tm    = blockIdx.x;            // M tile (M multiple of 16)
    const int col0  = blockIdx.y * 64;       // 64-wide N strip (Nn mult of 64)
    const int row   = tm * 16 + (lane & 15);
    const int colb  = col0 + (lane & 15);
    const int khalf = (lane >> 4) << 1;      // 0 or 2

    const float* __restrict__ arow = A + (size_t)row * K;
    v8f acc0 = {}, acc1 = {}, acc2 = {}, acc3 = {};
#pragma unroll 4
    for (int k = 0; k < K; k += 4) {
        v2f a;
        a[0] = arow[k + khalf];
        a[1] = arow[k + khalf + 1];
        if (k + 64 < K) __builtin_prefetch(arow + k + 64, 0, 0);
        const float* __restrict__ brow0 = B + (size_t)(k + khalf) * Nn + colb;
        const float* __restrict__ brow1 = brow0 + Nn;
        v2f b;
        b[0] = brow0[0];  b[1] = brow1[0];
        acc0 = __builtin_amdgcn_wmma_f32_16x16x4_f32(false, a, false, b,
                                                     (short)0, acc0, false, false);
        b[0] = brow0[16]; b[1] = brow1[16];
        acc1 = __builtin_amdgcn_wmma_f32_16x16x4_f32(false, a, false, b,
                                                     (short)0, acc1, false, false);
        b[0] = brow0[32]; b[1] = brow1[32];
        acc2 = __builtin_amdgcn_wmma_f32_16x16x4_f32(false, a, false, b,
                                                     (short)0, acc2, false, false);
        b[0] = brow0[48]; b[1] = brow1[48];
        acc3 = __builtin_amdgcn_wmma_f32_16x16x4_f32(false, a, false, b,
                                                     (short)0, acc3, false, false);
    }
    const int crow = tm * 16 + (lane >> 4) * 8;
#pragma unroll
    for (int v = 0; v < 8; ++v) {
        float* crp = C + (size_t)(crow + v) * Nn + colb;
        crp[0]  = acc0[v];
        crp[16] = acc1[v];
        crp[32] = acc2[v];
        crp[48] = acc3[v];
    }
}

// ---------------- per-node attention logits: al = h . a  ----------------
__global__ void logits_kernel(const float* __restrict__ h,
                              const float* __restrict__ a_src,
                              const float* __restrict__ a_dst,
                              float* __restrict__ al_s,
                              float* __restrict__ al_d,
                              int n_nodes, int H, int Cc) {
    int t = blockIdx.x * blockDim.x + threadIdx.x;
    if (t >= n_nodes * H) return;
    int n = t / H, hd = t % H;
    const float* hp = h + (size_t)n * H * Cc + (size_t)hd * Cc;
    const float* as = a_src + (size_t)hd * Cc;
    const float* ad = a_dst + (size_t)hd * Cc;
    float ss = 0.f, sd = 0.f;
    for (int c = 0; c < Cc; ++c) {
        float v = hp[c];
        ss += v * as[c];
        sd += v * ad[c];
    }
    al_s[t] = ss;
    al_d[t] = sd;
}

// ---------------- edge phase A: leaky-relu logits + segment max ----------------
__global__ void edge_max_kernel(const int* __restrict__ ei,
                                const float* __restrict__ al_s,
                                const float* __restrict__ al_d,
                                float* __restrict__ evals,
                                unsigned* __restrict__ mbuf,
                                int H) {
    long long t = (long long)blockIdx.x * blockDim.x + threadIdx.x;
    long long total = (long long)EN_ * H;
    if (t >= total) return;
    int e = (int)(t / H), hd = (int)(t % H);
    int src, dst;
    if (e < EE_) { src = ei[e]; dst = ei[EE_ + e]; }
    else         { src = dst = e - EE_; }
    float v = al_s[(size_t)src * H + hd] + al_d[(size_t)dst * H + hd];
    v = v > 0.f ? v : NEG_ * v;
    evals[(size_t)e * H + hd] = v;
    atomicMax(&mbuf[(size_t)dst * H + hd], fmap(v));
}

// ---------------- edge phase B: exp(e - max) + segment sum ----------------
__global__ void edge_exp_kernel(const int* __restrict__ ei,
                                const unsigned* __restrict__ mbuf,
                                float* __restrict__ evals,
                                float* __restrict__ sbuf,
                                int H) {
    long long t = (long long)blockIdx.x * blockDim.x + threadIdx.x;
    long long total = (long long)EN_ * H;
    if (t >= total) return;
    int e = (int)(t / H), hd = (int)(t % H);
    int dst = (e < EE_) ? ei[EE_ + e] : (e - EE_);
    float m = funmap(mbuf[(size_t)dst * H + hd]);
    float x = expf(evals[(size_t)e * H + hd] - m);
    evals[(size_t)e * H + hd] = x;
    atomicAdd(&sbuf[(size_t)dst * H + hd], x);
}

// ---------------- edge phase C: agg[dst] += alpha * h[src] ----------------
// one thread per (edge, 4-channel chunk); float4 gather + 4 atomic adds
__global__ void edge_agg_kernel(const int* __restrict__ ei,
                                const float* __restrict__ h,
                                const float* __restrict__ evals,
                                const float* __restrict__ sbuf,
                                float* __restrict__ agg,
                                int H, int Cc) {
    const int F = H * Cc;
    const int chunks = F >> 2;
    long long t = (long long)blockIdx.x * blockDim.x + threadIdx.x;
    long long total = (long long)EN_ * chunks;
    if (t >= total) return;
    int e = (int)(t / chunks);
    int cq = (int)(t % chunks) << 2;       // channel base within [0, F)
    int hd = cq / Cc;
    int src, dst;
    if (e < EE_) { src = ei[e]; dst = ei[EE_ + e]; }
    else         { src = dst = e - EE_; }
    float alpha = evals[(size_t)e * H + hd] /
                  (sbuf[(size_t)dst * H + hd] + 1e-16f);
    const float4 hv = *(const float4*)(h + (size_t)src * F + cq);
    float* out = agg + (size_t)dst * F + cq;
    atomicAdd(out + 0, hv.x * alpha);
    atomicAdd(out + 1, hv.y * alpha);
    atomicAdd(out + 2, hv.z * alpha);
    atomicAdd(out + 3, hv.w * alpha);
}

// ---------------- bias + ELU (in place ok) ----------------
__global__ void bias_elu_kernel(const float* __restrict__ in,
                                const float* __restrict__ bias,
                                float* __restrict__ out,
                                long long total, int F) {
    long long t = (long long)blockIdx.x * blockDim.x + threadIdx.x;
    if (t >= total) return;
    int ch = (int)(t % F);
    float v = in[t] + bias[ch];
    out[t] = v > 0.f ? v : expm1f(v);
}

// ---------------- global mean pool: accumulate sums & counts ----------------
__global__ void pool_sum_kernel(const float* __restrict__ h,
                                const int* __restrict__ batch,
                                float* __restrict__ sums,
                                float* __restrict__ cnt) {
    long long t = (long long)blockIdx.x * blockDim.x + threadIdx.x;
    long long total = (long long)NN_ * OUT_;
    if (t >= total) return;
    int n = (int)(t / OUT_);
    int ch = (int)(t % OUT_);
    int g = batch[n];
    atomicAdd(&sums[(size_t)g * OUT_ + ch], h[t]);
    if (ch == 0) atomicAdd(&cnt[g], 1.0f);
}

// ---------------- pooled @ fc_W + fc_b, ReLU ----------------
__global__ void fc_relu_kernel(const float* __restrict__ sums,
                               const float* __restrict__ cnt,
                               const float* __restrict__ W,
                               const float* __restrict__ b,
                               float* __restrict__ out) {
    int t = blockIdx.x * blockDim.x + threadIdx.x;
    if (t >= GG_ * OUT_) return;
    int g = t / OUT_, o = t % OUT_;
    float c = cnt[g];
    c = c > 1.f ? c : 1.f;
    float inv_c = 1.f / c;
    float acc = b[o];
    const float* sg = sums + (size_t)g * OUT_;
    for (int k = 0; k < OUT_; ++k)
        acc += (sg[k] * inv_c) * W[(size_t)k * OUT_ + o];
    out[t] = acc > 0.f ? acc : 0.f;
}

// ---------------- workspace layout ----------------
static const size_t SZ_BUF  = (size_t)NN_ * F1_ * 4;          // 51.2 MB
static const size_t SZ_AL   = (size_t)NN_ * HEADS_ * 4;       // 0.8 MB
static const size_t SZ_EX   = (size_t)EN_ * HEADS_ * 4;       // 13.6 MB
static const size_t OFF_A   = 0;                              // h1 | h2+agg2
static const size_t OFF_B   = OFF_A + SZ_BUF;                 // agg1 -> x2
static const size_t OFF_ALS = OFF_B + SZ_BUF;
static const size_t OFF_ALD = OFF_ALS + SZ_AL;
static const size_t OFF_M   = OFF_ALD + SZ_AL;
static const size_t OFF_S   = OFF_M + SZ_AL;
static const size_t OFF_EX  = OFF_S + SZ_AL;
static const size_t OFF_SUM = OFF_EX + SZ_EX;
static const size_t OFF_CNT = OFF_SUM + (size_t)GG_ * OUT_ * 4;

static inline int blocks_for(long long total, int bs) {
    return (int)((total + bs - 1) / bs);
}

extern "C" void kernel_launch(void* const* d_in, const int* in_sizes, int n_in,
                              void* d_out, int out_size, void* d_ws, size_t ws_size,
                              hipStream_t stream) {
    (void)in_sizes; (void)n_in; (void)out_size; (void)ws_size;
    const float* x    = (const float*)d_in[0];
    const int*   ei   = (const int*)d_in[1];
    const int*   batch= (const int*)d_in[2];
    const float* W1   = (const float*)d_in[3];
    const float* a1s  = (const float*)d_in[4];
    const float* a1d  = (const float*)d_in[5];
    const float* b1   = (const float*)d_in[6];
    const float* W2   = (const float*)d_in[7];
    const float* a2s  = (const float*)d_in[8];
    const float* a2d  = (const float*)d_in[9];
    const float* b2   = (const float*)d_in[10];
    const float* fcW  = (const float*)d_in[11];
    const float* fcb  = (const float*)d_in[12];
    float* out = (float*)d_out;

    char* ws = (char*)d_ws;
    float*    h1   = (float*)(ws + OFF_A);
    float*    agg1 = (float*)(ws + OFF_B);   // becomes x2 after ELU
    float*    als  = (float*)(ws + OFF_ALS);
    float*    ald  = (float*)(ws + OFF_ALD);
    unsigned* mbuf = (unsigned*)(ws + OFF_M);
    float*    sbuf = (float*)(ws + OFF_S);
    float*    ex   = (float*)(ws + OFF_EX);
    float*    sums = (float*)(ws + OFF_SUM);
    float*    cnt  = (float*)(ws + OFF_CNT);
    float*    h2   = (float*)(ws + OFF_A);                              // reuse h1
    float*    agg2 = (float*)(ws + OFF_A + (size_t)NN_ * OUT_ * 4);     // upper half

    const int BS = 256;

    // ===== layer 1 =====
    hipMemsetAsync(agg1, 0, SZ_BUF, stream);
    hipMemsetAsync(ws + OFF_M, 0, 2 * SZ_AL, stream);   // mbuf + sbuf

    wmma_gemm_f32<<<dim3(NN_ / 16, F1_ / 64), 32, 0, stream>>>(x, W1, h1, NN_, IN_, F1_);

    logits_kernel<<<blocks_for((long long)NN_ * HEADS_, BS), BS, 0, stream>>>(
        h1, a1s, a1d, als, ald, NN_, HEADS_, HID_);

    edge_max_kernel<<<blocks_for((long long)EN_ * HEADS_, BS), BS, 0, stream>>>(
        ei, als, ald, ex, mbuf, HEADS_);
    edge_exp_kernel<<<blocks_for((long long)EN_ * HEADS_, BS), BS, 0, stream>>>(
        ei, mbuf, ex, sbuf, HEADS_);
    edge_agg_kernel<<<blocks_for((long long)EN_ * (F1_ / 4), BS), BS, 0, stream>>>(
        ei, h1, ex, sbuf, agg1, HEADS_, HID_);

    bias_elu_kernel<<<blocks_for((long long)NN_ * F1_, BS), BS, 0, stream>>>(
        agg1, b1, agg1, (long long)NN_ * F1_, F1_);     // agg1 -> x2 in place

    // ===== layer 2 (reuses small buffers; stream-ordered) =====
    hipMemsetAsync(ws + OFF_M, 0, 2 * SZ_AL, stream);                   // mbuf + sbuf
    hipMemsetAsync(agg2, 0, (size_t)NN_ * OUT_ * 4, stream);
    hipMemsetAsync(ws + OFF_SUM, 0, (size_t)GG_ * OUT_ * 4 + 256, stream); // sums + cnt

    wmma_gemm_f32<<<dim3(NN_ / 16, OUT_ / 64), 32, 0, stream>>>(agg1, W2, h2, NN_, F1_, OUT_);

    logits_kernel<<<blocks_for((long long)NN_, BS), BS, 0, stream>>>(
        h2, a2s, a2d, als, ald, NN_, 1, OUT_);

    edge_max_kernel<<<blocks_for((long long)EN_, BS), BS, 0, stream>>>(
        ei, als, ald, ex, mbuf, 1);
    edge_exp_kernel<<<blocks_for((long long)EN_, BS), BS, 0, stream>>>(
        ei, mbuf, ex, sbuf, 1);
    edge_agg_kernel<<<blocks_for((long long)EN_ * (OUT_ / 4), BS), BS, 0, stream>>>(
        ei, h2, ex, sbuf, agg2, 1, OUT_);

    bias_elu_kernel<<<blocks_for((long long)NN_ * OUT_, BS), BS, 0, stream>>>(
        agg2, b2, agg2, (long long)NN_ * OUT_, OUT_);

    // ===== pool + fc =====
    pool_sum_kernel<<<blocks_for((long long)NN_ * OUT_, BS), BS, 0, stream>>>(
        agg2, batch, sums, cnt);
    fc_relu_kernel<<<blocks_for((long long)GG_ * OUT_, BS), BS, 0, stream>>>(
        sums, cnt, fcW, fcb, out);
}